// adder2d_65128884077246
// MI455X (gfx1250) — compile-verified
//
#include <hip/hip_runtime.h>
#include <stdint.h>

#define NB    8
#define CH    32
#define HH    32
#define WW    32
#define FF    32
#define PADH  34
#define QXP_BYTES (NB*PADH*PADH*CH)   // 295936 per array
#define QW_BYTES  (3*3*FF*8*4)        // 9216 per array
#define QW_DWORDS (3*3*FF*8)          // 2304 dwords per array
#define ZFILL_DW  (2*QXP_BYTES/4)     // 147968 dwords (both masked x arrays, contiguous)

// ---------------- prep 0: zero both padded masked-x arrays (borders become 0) ----------------
__global__ __launch_bounds__(256) void zerofill_kernel(uint32_t* __restrict__ p) {
    int tid = blockIdx.x * 256 + threadIdx.x;
    p[tid] = 0u;   // grid sized exactly to ZFILL_DW
}

// ---------------- prep 1: quantize interior of x into masked halves [n][y][x][c] u8 ----------------
// x is [n][c][y][xw] linear; tid maps onto it directly -> fully coalesced reads.
__global__ __launch_bounds__(256) void quantx_kernel(const float* __restrict__ x,
                                                     uint8_t* __restrict__ qxp7,
                                                     uint8_t* __restrict__ qxp8) {
    int tid = blockIdx.x * 256 + threadIdx.x;        // 262144 threads
    int xw = tid & 31;
    int y  = (tid >> 5) & 31;
    int c  = (tid >> 10) & 31;
    int n  = tid >> 15;
    float v = x[tid] * 256.0f;
    v = fminf(fmaxf(v, 0.0f), 255.0f);
    uint32_t q = (uint32_t)(int)v;                   // truncation toward zero == astype(int32)
    int idx = (((n*PADH) + (y + 1))*PADH + (xw + 1))*CH + c;
    qxp7[idx] = (uint8_t)(q & 0x7fu);
    qxp8[idx] = (uint8_t)(q & 0x80u);
}

// ---------------- prep 2: quantize W into dword-packed masked halves [kh][kw][f][cb] ----------------
__global__ __launch_bounds__(256) void quantw_kernel(const float* __restrict__ w,
                                                     uint8_t* __restrict__ qwt7,
                                                     uint8_t* __restrict__ qwt8) {
    int tid = blockIdx.x * 256 + threadIdx.x;
    if (tid >= FF*CH*9) return;
    // W layout: [f][c][kh][kw], kw fastest
    int kw = tid % 3;  int t = tid / 3;
    int kh = t % 3;    t /= 3;
    int c  = t % CH;   int f = t / CH;
    float v = w[tid] * 256.0f;
    v = fminf(fmaxf(v, 0.0f), 255.0f);
    uint32_t q = (uint32_t)(int)v;
    int dst = ((((kh*3 + kw)*FF + f)*8 + (c >> 2)) << 2) + (c & 3);
    qwt7[dst] = (uint8_t)(q & 0x7fu);
    qwt8[dst] = (uint8_t)(q & 0x80u);
}

// ---------------- main: per-output mod-256 sum via add + v_xor3 + v_dot4_u32_u8 ----------------
__global__ __launch_bounds__(256) void adder_main_kernel(const uint8_t* __restrict__ qxp7,
                                                         const uint8_t* __restrict__ qxp8,
                                                         const uint32_t* __restrict__ qwt7,
                                                         const uint32_t* __restrict__ qwt8,
                                                         float* __restrict__ out) {
    __shared__ uint32_t ldsA7[QW_DWORDS];
    __shared__ uint32_t ldsA8[QW_DWORDS];
    const int tid = threadIdx.x;

    // CDNA5 async global->LDS staging of both masked weight halves (ASYNCcnt-tracked).
    {
        uint32_t lb7 = (uint32_t)(uintptr_t)(&ldsA7[0]);
        uint32_t lb8 = (uint32_t)(uintptr_t)(&ldsA8[0]);
        #pragma unroll
        for (int r = 0; r < 9; ++r) {
            int idx = tid + r * 256;
            uint32_t lofs7 = lb7 + ((uint32_t)idx << 2);
            uint32_t lofs8 = lb8 + ((uint32_t)idx << 2);
            uint64_t ga7 = (uint64_t)(uintptr_t)(qwt7 + idx);
            uint64_t ga8 = (uint64_t)(uintptr_t)(qwt8 + idx);
            asm volatile("global_load_async_to_lds_b32 %0, %1, off"
                         :: "v"(lofs7), "v"(ga7) : "memory");
            asm volatile("global_load_async_to_lds_b32 %0, %1, off"
                         :: "v"(lofs8), "v"(ga8) : "memory");
        }
        asm volatile("s_wait_asynccnt 0x0" ::: "memory");
    }
    __syncthreads();

    const int j  = tid & 31;         // lane -> output column (coalesced stores)
    const int fo = tid >> 5;         // wave -> filter within chunk (wave-uniform)
    int b = blockIdx.x;
    const int fchunk = b & 3;  b >>= 2;
    const int i      = b & 31; b >>= 5;
    const int n      = b;            // 0..7
    const int f      = fchunk * 8 + fo;

    uint32_t acc0 = 0, acc1 = 0;
    const uint32_t ones = 0x01010101u;

    #pragma unroll
    for (int kh = 0; kh < 3; ++kh) {
        const int rowoff = (((n*PADH) + (i + kh))*PADH + j) * CH;
        #pragma unroll
        for (int kw = 0; kw < 3; ++kw) {
            const uint32_t* b7p = (const uint32_t*)(qxp7 + rowoff + kw*CH);  // 8 dwords, 16B aligned
            const uint32_t* b8p = (const uint32_t*)(qxp8 + rowoff + kw*CH);
            const uint32_t* a7p = &ldsA7[((kh*3 + kw)*FF + f) * 8];          // wave-broadcast
            const uint32_t* a8p = &ldsA8[((kh*3 + kw)*FF + f) * 8];
            #pragma unroll
            for (int cb = 0; cb < 8; cb += 2) {
                uint32_t t0 = a7p[cb]     + b7p[cb];          // no cross-byte carry (<=254/byte)
                uint32_t t1 = a7p[cb + 1] + b7p[cb + 1];
                uint32_t s0 = t0 ^ a8p[cb]     ^ b8p[cb];     // v_xor3_b32: (a+b) mod 256 per byte
                uint32_t s1 = t1 ^ a8p[cb + 1] ^ b8p[cb + 1];
                asm volatile("v_dot4_u32_u8 %0, %1, %2, %0" : "+v"(acc0) : "v"(s0), "v"(ones));
                asm volatile("v_dot4_u32_u8 %0, %1, %2, %0" : "+v"(acc1) : "v"(s1), "v"(ones));
            }
        }
    }
    uint32_t acc = acc0 + acc1;
    out[((n*FF + f)*HH + i)*WW + j] = -(float)acc * (1.0f / 256.0f);
}

extern "C" void kernel_launch(void* const* d_in, const int* in_sizes, int n_in,
                              void* d_out, int out_size, void* d_ws, size_t ws_size,
                              hipStream_t stream) {
    const float* x = (const float*)d_in[0];   // [8,32,32,32]
    const float* w = (const float*)d_in[1];   // [32,32,3,3]
    float* out = (float*)d_out;               // [8,32,32,32]

    uint8_t* qxp7 = (uint8_t*)d_ws;                                   // 295936 B
    uint8_t* qxp8 = qxp7 + QXP_BYTES;                                 // 295936 B (contiguous with qxp7)
    uint8_t* qwt7 = qxp8 + QXP_BYTES;                                 // 9216 B (4B aligned)
    uint8_t* qwt8 = qwt7 + QW_BYTES;                                  // 9216 B

    zerofill_kernel<<<ZFILL_DW / 256, 256, 0, stream>>>((uint32_t*)d_ws);
    quantx_kernel<<<(NB*CH*HH*WW) / 256, 256, 0, stream>>>(x, qxp7, qxp8);
    quantw_kernel<<<(FF*CH*9) / 256, 256, 0, stream>>>(w, qwt7, qwt8);
    adder_main_kernel<<<NB*HH*4, 256, 0, stream>>>(qxp7, qxp8,
                                                   (const uint32_t*)qwt7,
                                                   (const uint32_t*)qwt8, out);
}